// DoubleHashEmbedder_40716289966787
// MI455X (gfx1250) — compile-verified
//
#include <hip/hip_runtime.h>

#define NLV   16
#define L2_3D 15
#define L2_4D 22
#define BPTS  262144

#define PR1 2654435761u
#define PR2 805459861u
#define PR3 3674653429u

// RES  = floor(16 * 2^(l/3)),  RES_T = floor(2 * 2^(4l/15))  (precomputed, matches numpy)
static __constant__ float c_res[NLV]  = {16.f,20.f,25.f,32.f,40.f,50.f,64.f,80.f,
                                         101.f,128.f,161.f,203.f,256.f,322.f,406.f,512.f};
static __constant__ float c_rest[NLV] = {2.f,2.f,2.f,3.f,4.f,5.f,6.f,7.f,
                                         8.f,10.f,12.f,15.f,18.f,22.f,26.f,32.f};

#if defined(__gfx1250__) && __has_builtin(__builtin_amdgcn_global_load_async_to_lds_b128)
#define HAVE_ASYNC 1
#else
#define HAVE_ASYNC 0
#endif

#if HAVE_ASYNC
typedef int v4i __attribute__((ext_vector_type(4)));
typedef v4i __attribute__((address_space(1))) *gptr_v4i;  // global
typedef v4i __attribute__((address_space(3))) *lptr_v4i;  // LDS
#endif

struct __attribute__((aligned(8))) F2 { float x, y; };

__device__ __forceinline__ F2 lerp2(F2 a, F2 b, float w) {
    const float om = 1.0f - w;
    F2 r;
    r.x = fmaf(om, a.x, w * b.x);
    r.y = fmaf(om, a.y, w * b.y);
    return r;
}

// Stage one 256KB level table (32768 x float2) of emb3d into LDS.
// 1024 threads x 16 iters x 16B = 262144 bytes, fully coalesced.
__device__ __forceinline__ void stage_level(const float* __restrict__ emb3d,
                                            float* __restrict__ s_tab,
                                            int lvl, int tid)
{
    const char* gsrc = (const char*)(emb3d + ((size_t)lvl << (L2_3D + 1)));
    char*       ldst = (char*)s_tab;
#pragma unroll
    for (int i = 0; i < 16; ++i) {
        const int off = (i * 1024 + tid) * 16;
#if HAVE_ASYNC
        __builtin_amdgcn_global_load_async_to_lds_b128(
            (gptr_v4i)(gsrc + off),
            (lptr_v4i)(ldst + off),
            0, 0);
#else
        float4 t = *(const float4*)(gsrc + off);
        *(float4*)(ldst + off) = t;
#endif
    }
}

__device__ __forceinline__ void wait_async_done() {
#if HAVE_ASYNC
#if __has_builtin(__builtin_amdgcn_s_wait_asynccnt)
    __builtin_amdgcn_s_wait_asynccnt(0);
#else
    asm volatile("s_wait_asynccnt 0" ::: "memory");
#endif
#endif
}

extern "C" __global__ void __launch_bounds__(1024)
hash_embed_kernel(const float* __restrict__ x,
                  const float* __restrict__ emb4d,
                  const float* __restrict__ emb3d,
                  float* __restrict__ out)
{
    extern __shared__ float s_tab[];   // 32768 * float2 = 256 KB, set at launch
    const int tid = (int)threadIdx.x;
    const int p   = (int)blockIdx.x * 1024 + tid;

    const float4 xv = ((const float4*)x)[p];
    const float x0 = xv.x, x1 = xv.y, x2 = xv.z, x3 = xv.w;
    const float xc0 = fminf(fmaxf(x0, 0.f), 1.f);
    const float xc1 = fminf(fmaxf(x1, 0.f), 1.f);
    const float xc2 = fminf(fmaxf(x2, 0.f), 1.f);
    const float xc3 = fminf(fmaxf(x3, 0.f), 1.f);

    stage_level(emb3d, s_tab, 0, tid);   // prologue: level 0 table -> LDS

    for (int l = 0; l < NLV; ++l) {
        const float res = c_res[l];
        const float rt  = c_rest[l];

        const float bl0 = floorf(xc0 * res);
        const float bl1 = floorf(xc1 * res);
        const float bl2 = floorf(xc2 * res);
        const float bl3 = floorf(xc3 * rt);
        const float w0 = fmaf(x0, res, -bl0);
        const float w1 = fmaf(x1, res, -bl1);
        const float w2 = fmaf(x2, res, -bl2);
        const float w3 = fmaf(x3, rt,  -bl3);

        const unsigned i0 = (unsigned)(int)bl0;
        const unsigned i1 = (unsigned)(int)bl1;
        const unsigned i2 = (unsigned)(int)bl2;
        const unsigned i3 = (unsigned)(int)bl3;

        // per-dimension hash partials (corner offsets 0/1)
        const unsigned a0 = i0,            a1 = i0 + 1u;
        const unsigned b0 = i1 * PR1,      b1 = (i1 + 1u) * PR1;
        const unsigned g0 = i2 * PR2,      g1 = (i2 + 1u) * PR2;
        const unsigned d0 = i3 * PR3,      d1 = (i3 + 1u) * PR3;

        // ---- 4D level: 16 hashed global gathers (issue all -> max MLP) ----
        const F2* __restrict__ T4 = (const F2*)emb4d + ((size_t)l << L2_4D);
        F2 v[16];
#pragma unroll
        for (int c = 0; c < 16; ++c) {
            unsigned h = ((c & 8) ? a1 : a0) ^ ((c & 4) ? b1 : b0)
                       ^ ((c & 2) ? g1 : g0) ^ ((c & 1) ? d1 : d0);
            h &= (1u << L2_4D) - 1u;
            v[c] = T4[h];                       // global_load_b64
        }
#pragma unroll
        for (int c = 0; c < 8; ++c) v[c] = lerp2(v[c], v[c + 8], w0);
#pragma unroll
        for (int c = 0; c < 4; ++c) v[c] = lerp2(v[c], v[c + 4], w1);
        v[0] = lerp2(v[0], v[2], w2);
        v[1] = lerp2(v[1], v[3], w2);
        const F2 e4 = lerp2(v[0], v[1], w3);

        // ---- staging of this level's 3D table is complete for all waves ----
        wait_async_done();
        __syncthreads();

        // ---- 3D level: 8 hashed LDS gathers ----
        const F2* __restrict__ st = (const F2*)s_tab;
        F2 u[8];
#pragma unroll
        for (int c = 0; c < 8; ++c) {
            unsigned h = ((c & 4) ? a1 : a0) ^ ((c & 2) ? b1 : b0) ^ ((c & 1) ? g1 : g0);
            h &= (1u << L2_3D) - 1u;
            u[c] = st[h];                       // ds_load_b64
        }
#pragma unroll
        for (int c = 0; c < 4; ++c) u[c] = lerp2(u[c], u[c + 4], w0);
        u[0] = lerp2(u[0], u[2], w1);
        u[1] = lerp2(u[1], u[3], w1);
        const F2 e3 = lerp2(u[0], u[1], w2);

        __syncthreads();                        // everyone done reading LDS level l
        if (l + 1 < NLV)
            stage_level(emb3d, s_tab, l + 1, tid);  // async DMA overlaps next 4D gathers

        F2 o;
        o.x = e4.x + e3.x;
        o.y = e4.y + e3.y;
        *(F2*)(out + (size_t)p * 32 + 2 * l) = o;   // global_store_b64
    }

    // keep = keep4 & keep3  (keep3 implied by first 3 dims of keep4)
    const bool keep = (x0 >= 0.f) & (x0 <= 1.f) & (x1 >= 0.f) & (x1 <= 1.f)
                    & (x2 >= 0.f) & (x2 <= 1.f) & (x3 >= 0.f) & (x3 <= 1.f);
    out[(size_t)BPTS * 32 + p] = keep ? 1.0f : 0.0f;
}

extern "C" void kernel_launch(void* const* d_in, const int* in_sizes, int n_in,
                              void* d_out, int out_size, void* d_ws, size_t ws_size,
                              hipStream_t stream)
{
    const float* x     = (const float*)d_in[0];
    const float* emb4d = (const float*)d_in[1];
    const float* emb3d = (const float*)d_in[2];
    float* out = (float*)d_out;

    const size_t shmem = (size_t)(1u << L2_3D) * 2u * sizeof(float);  // 262144 B
    // opt in to >64KB dynamic LDS (no-op if driver already allows it)
    (void)hipFuncSetAttribute((const void*)hash_embed_kernel,
                              hipFuncAttributeMaxDynamicSharedMemorySize,
                              (int)shmem);

    dim3 grid(BPTS / 1024), block(1024);
    hash_embed_kernel<<<grid, block, shmem, stream>>>(x, emb4d, emb3d, out);
}